// Net_30296699306334
// MI455X (gfx1250) — compile-verified
//
#include <hip/hip_runtime.h>
#include <hip/hip_bf16.h>

typedef __attribute__((ext_vector_type(2))) float v2f;
typedef __attribute__((ext_vector_type(8))) float v8f;

// ---------------------------------------------------------------------------
// Degree / normalization
// ---------------------------------------------------------------------------
__global__ void init_deg_kernel(float* __restrict__ deg, int N) {
    int i = blockIdx.x * blockDim.x + threadIdx.x;
    if (i < N) deg[i] = 1.0f;  // self-loop contributes 1
}

__global__ void count_deg_kernel(const int* __restrict__ dst, float* __restrict__ deg, int E) {
    int e = blockIdx.x * blockDim.x + threadIdx.x;
    if (e < E) atomicAdd(&deg[dst[e]], 1.0f);
}

__global__ void dinv_kernel(const float* __restrict__ deg, float* __restrict__ dinv, int N) {
    int i = blockIdx.x * blockDim.x + threadIdx.x;
    if (i < N) dinv[i] = rsqrtf(deg[i]);
}

// ---------------------------------------------------------------------------
// Dense GEMM via V_WMMA_F32_16X16X4_F32 (wave32).
//   Out[M x (NT*16)] = op(A)[M x K] @ W[K x (NT*16)]
// One wave per 16-row M tile; each wave owns all NT column tiles so the A
// fragment is reused NT times per k-step.
//
// Fragment layouts (ISA 7.12.2):
//   A (16x4 f32):  lanes 0-15 rows 0-15 / K={0,1}; lanes 16-31 rows / K={2,3}
//   B (4x16 f32):  lanes 0-15 cols / K={0,1};      lanes 16-31 cols / K={2,3}
//   C/D (16x16 f32): VGPR v: lanes 0-15 -> M=v, N=lane; lanes 16-31 -> M=v+8
// ---------------------------------------------------------------------------
template <int NT, int K, bool RELU_A>
__global__ __launch_bounds__(256) void gemm_wmma_f32_kernel(
    const float* __restrict__ A, const float* __restrict__ W,
    float* __restrict__ Out, int M) {
    const int NCOLS = NT * 16;
    const int wave  = threadIdx.x >> 5;
    const int lane  = threadIdx.x & 31;
    const int tileM = blockIdx.x * 8 + wave;
    if (tileM * 16 >= M) return;  // wave-uniform: EXEC stays all-ones for WMMA

    const int rowA  = tileM * 16 + (lane & 15);
    const int khalf = (lane >> 4) * 2;  // 0 or 2
    const int col   = lane & 15;

    v8f acc[NT];
#pragma unroll
    for (int t = 0; t < NT; ++t)
#pragma unroll
        for (int v = 0; v < 8; ++v) acc[t][v] = 0.0f;

    for (int k = 0; k < K; k += 4) {
        float a0 = A[(long long)rowA * K + k + khalf];
        float a1 = A[(long long)rowA * K + k + khalf + 1];
        if (RELU_A) { a0 = fmaxf(a0, 0.0f); a1 = fmaxf(a1, 0.0f); }
        v2f a; a.x = a0; a.y = a1;
#pragma unroll
        for (int t = 0; t < NT; ++t) {
            v2f b;
            b.x = W[(k + khalf) * NCOLS + t * 16 + col];
            b.y = W[(k + khalf + 1) * NCOLS + t * 16 + col];
            acc[t] = __builtin_amdgcn_wmma_f32_16x16x4_f32(
                false, a, false, b, (short)0, acc[t], false, false);
        }
    }

#pragma unroll
    for (int t = 0; t < NT; ++t) {
#pragma unroll
        for (int v = 0; v < 8; ++v) {
            int r = tileM * 16 + v + (lane >> 4) * 8;
            Out[(long long)r * NCOLS + t * 16 + col] = acc[t][v];
        }
    }
}

// ---------------------------------------------------------------------------
// agg[n][c] = dinv[n]^2 * h[n][c] + bias[c]   (self-loop + bias init)
// ---------------------------------------------------------------------------
template <int C>
__global__ void init_self_kernel(const float* __restrict__ h,
                                 const float* __restrict__ dinv,
                                 const float* __restrict__ bias,
                                 float* __restrict__ agg, int N) {
    long long i = (long long)blockIdx.x * blockDim.x + threadIdx.x;
    if (i >= (long long)N * C) return;
    int n = (int)(i / C);
    int c = (int)(i % C);
    float di = dinv[n];
    agg[i] = di * di * h[i] + bias[c];
}

// ---------------------------------------------------------------------------
// agg[dst] += dinv[src]*dinv[dst] * h[src]   (CH4 = channels/4 lanes per edge)
// ---------------------------------------------------------------------------
template <int CH4>
__global__ void edge_agg_kernel(const int* __restrict__ src,
                                const int* __restrict__ dst,
                                const float* __restrict__ dinv,
                                const float* __restrict__ h,
                                float* __restrict__ agg, int E) {
    long long tid = (long long)blockIdx.x * blockDim.x + threadIdx.x;
    int e  = (int)(tid / CH4);
    int c4 = (int)(tid % CH4);
    if (e >= E) return;
    int s = src[e], d = dst[e];
    float norm = dinv[s] * dinv[d];
    const float4* h4 = (const float4*)h;
    float4 hv = h4[(long long)s * CH4 + c4];
    float* ap = agg + (long long)d * (CH4 * 4) + c4 * 4;
    atomicAdd(ap + 0, norm * hv.x);
    atomicAdd(ap + 1, norm * hv.y);
    atomicAdd(ap + 2, norm * hv.z);
    atomicAdd(ap + 3, norm * hv.w);
}

// ---------------------------------------------------------------------------
// out[l] = dot(z[a_l], z[b_l]) over 64 channels; one wave per label edge.
// ---------------------------------------------------------------------------
__global__ void decode_kernel(const float* __restrict__ z,
                              const int* __restrict__ eli,
                              float* __restrict__ out, int L) {
    long long gid = (long long)blockIdx.x * blockDim.x + threadIdx.x;
    int w    = (int)(gid >> 5);
    int lane = (int)(gid & 31);
    if (w >= L) return;
    int a = eli[w];
    int b = eli[L + w];
    const float2* za = (const float2*)(z + (long long)a * 64);
    const float2* zb = (const float2*)(z + (long long)b * 64);
    float2 pa = za[lane];
    float2 pb = zb[lane];
    float p = pa.x * pb.x + pa.y * pb.y;
    p += __shfl_xor(p, 16, 32);
    p += __shfl_xor(p, 8, 32);
    p += __shfl_xor(p, 4, 32);
    p += __shfl_xor(p, 2, 32);
    p += __shfl_xor(p, 1, 32);
    if (lane == 0) out[w] = p;
}

// ---------------------------------------------------------------------------
// Launch
// ---------------------------------------------------------------------------
static inline size_t align256(size_t x) { return (x + 255) & ~size_t(255); }

extern "C" void kernel_launch(void* const* d_in, const int* in_sizes, int n_in,
                              void* d_out, int out_size, void* d_ws, size_t ws_size,
                              hipStream_t stream) {
    const int IN = 256, HID = 128, OUT = 64;
    const int N = in_sizes[0] / IN;
    const int E = in_sizes[5] / 2;
    const int L = in_sizes[6] / 2;

    const float* x   = (const float*)d_in[0];
    const float* W1  = (const float*)d_in[1];
    const float* b1  = (const float*)d_in[2];
    const float* W2  = (const float*)d_in[3];
    const float* b2  = (const float*)d_in[4];
    const int*   ei  = (const int*)d_in[5];
    const int*   eli = (const int*)d_in[6];
    const int* src = ei;
    const int* dst = ei + E;
    float* out = (float*)d_out;

    // Workspace carve-out
    char* ws = (char*)d_ws;
    size_t off = 0;
    float* deg  = (float*)(ws + off); off = align256(off + (size_t)N * 4);
    float* dinv = (float*)(ws + off); off = align256(off + (size_t)N * 4);
    float* h1   = (float*)(ws + off); off = align256(off + (size_t)N * HID * 4);
    float* agg1 = (float*)(ws + off); off = align256(off + (size_t)N * HID * 4);
    float* h2   = (float*)(ws + off); off = align256(off + (size_t)N * OUT * 4);
    float* z    = (float*)(ws + off); off = align256(off + (size_t)N * OUT * 4);
    (void)ws_size;

    const int TB = 256;

    // 1) degrees + dinv
    init_deg_kernel<<<(N + TB - 1) / TB, TB, 0, stream>>>(deg, N);
    count_deg_kernel<<<(E + TB - 1) / TB, TB, 0, stream>>>(dst, deg, E);
    dinv_kernel<<<(N + TB - 1) / TB, TB, 0, stream>>>(deg, dinv, N);

    // 2) GEMM1: h1 = x @ W1   [N x 128], K=256, 8 column tiles
    {
        int mtiles = (N + 15) / 16;
        int blocks = (mtiles + 7) / 8;
        gemm_wmma_f32_kernel<8, 256, false><<<blocks, TB, 0, stream>>>(x, W1, h1, N);
    }

    // 3) agg1 = dinv^2*h1 + b1, then scatter edges (32 lanes/edge, float4 each)
    {
        long long tot = (long long)N * HID;
        init_self_kernel<128><<<(unsigned)((tot + TB - 1) / TB), TB, 0, stream>>>(h1, dinv, b1, agg1, N);
        long long et = (long long)E * (HID / 4);
        edge_agg_kernel<32><<<(unsigned)((et + TB - 1) / TB), TB, 0, stream>>>(src, dst, dinv, h1, agg1, E);
    }

    // 4) GEMM2: h2 = relu(agg1) @ W2   [N x 64], K=128, 4 column tiles
    {
        int mtiles = (N + 15) / 16;
        int blocks = (mtiles + 7) / 8;
        gemm_wmma_f32_kernel<4, 128, true><<<blocks, TB, 0, stream>>>(agg1, W2, h2, N);
    }

    // 5) z = dinv^2*h2 + b2, then scatter edges (16 lanes/edge)
    {
        long long tot = (long long)N * OUT;
        init_self_kernel<64><<<(unsigned)((tot + TB - 1) / TB), TB, 0, stream>>>(h2, dinv, b2, z, N);
        long long et = (long long)E * (OUT / 4);
        edge_agg_kernel<16><<<(unsigned)((et + TB - 1) / TB), TB, 0, stream>>>(src, dst, dinv, h2, z, E);
    }

    // 6) decode: one wave per label edge
    {
        long long tt = (long long)L * 32;
        decode_kernel<<<(unsigned)((tt + TB - 1) / TB), TB, 0, stream>>>(z, eli, out, L);
    }
    (void)n_in; (void)out_size;
}